// GeneralIsotropicHardening_13640816132174
// MI455X (gfx1250) — compile-verified
//
#include <hip/hip_runtime.h>
#include <math.h>

typedef __attribute__((ext_vector_type(2))) float v2f;
typedef __attribute__((ext_vector_type(8))) float v8f;

namespace {
constexpr float kE     = 200000.0f;
constexpr float kNu    = 0.3f;
constexpr float kSig0  = 250.0f;
constexpr float kH     = 1000.0f;
constexpr float kMu    = kE / (2.0f * (1.0f + kNu));
constexpr float kLam   = kE * kNu / ((1.0f + kNu) * (1.0f - 2.0f * kNu));
constexpr float kTiny  = 1e-12f;
constexpr int   kIters = 16;
constexpr int   kThreads = 256;
constexpr int   kWaves   = kThreads / 32;
// per-wave LDS float offsets (32 points * 6 comps = 192 floats per array)
constexpr int O_EPS  = 0;
constexpr int O_EPSO = 192;
constexpr int O_SIGO = 384;
constexpr int O_POLD = 576;   // 32 floats
constexpr int O_ST   = 608;   // sig_trial, 192 floats
constexpr int WAVE_F = 800;
}

// ---- CDNA5 async global->LDS staging (guarded; falls back to plain copy) ----
#if defined(__AMDGCN__) && __has_builtin(__builtin_amdgcn_global_load_async_to_lds_b32)
#define ASYNC_B32(gp, lp)                                                      \
  __builtin_amdgcn_global_load_async_to_lds_b32((int *)(gp), (int *)(lp), 0, 0)
#else
#define ASYNC_B32(gp, lp) (*(lp) = *(gp))
#endif

#if defined(__AMDGCN__) && __has_builtin(__builtin_amdgcn_s_wait_asynccnt)
#define ASYNC_WAIT() __builtin_amdgcn_s_wait_asynccnt(0)
#elif defined(__AMDGCN__)
#define ASYNC_WAIT() asm volatile("s_wait_asynccnt 0" ::: "memory")
#else
#define ASYNC_WAIT() ((void)0)
#endif

// ---- FP32 WMMA (precision-preserving) for the elastic predictor tile ----
#if defined(__gfx1250__) && __has_builtin(__builtin_amdgcn_wmma_f32_16x16x4_f32)
#define HAVE_WMMA_F32X4 1
#else
#define HAVE_WMMA_F32X4 0
#endif

// 6x6 isotropic elastic stiffness in Voigt form, zero-padded to 16 columns.
__device__ __forceinline__ float cmat(int i, int j) {
  if (j >= 6) return 0.0f;
  float v = (i == j) ? 2.0f * kMu : 0.0f;
  if (i < 3 && j < 3) v += kLam;
  return v;
}

__global__ __launch_bounds__(kThreads)
void plasticity_return_map(const float* __restrict__ eps,
                           const float* __restrict__ eps_old,
                           const float* __restrict__ sig_old,
                           const float* __restrict__ p_old,
                           const float* __restrict__ epsp_old,
                           float* __restrict__ out_sig,
                           float* __restrict__ out_p,
                           float* __restrict__ out_epsp,
                           int N) {
  __shared__ float lds[kWaves * WAVE_F];
  const int tid  = threadIdx.x;
  const int wave = tid >> 5;
  const int lane = tid & 31;
  float* wl = &lds[wave * WAVE_F];

  const int ptBase = (blockIdx.x * kWaves + wave) * 32;  // first point of this wave
  const int pt6    = ptBase * 6;
  const int maxf6  = N * 6 - 1;
  const int maxp   = N - 1;

  // ---- Stage eps / eps_old / sig_old / p_old contiguously into LDS ----
  // in step k each lane copies float pt6 + k*32 + lane (clamped to the last) to k*32 + lane of its wave's LDS array, flat row-major (point,comp)
#pragma unroll
  for (int k = 0; k < 6; ++k) {
    int fi = pt6 + k * 32 + lane; if (fi > maxf6) fi = maxf6;
    ASYNC_B32(eps + fi, wl + O_EPS + k * 32 + lane);
  }
#pragma unroll
  for (int k = 0; k < 6; ++k) {
    int fi = pt6 + k * 32 + lane; if (fi > maxf6) fi = maxf6;
    ASYNC_B32(eps_old + fi, wl + O_EPSO + k * 32 + lane);
  }
#pragma unroll
  for (int k = 0; k < 6; ++k) {
    int fi = pt6 + k * 32 + lane; if (fi > maxf6) fi = maxf6;
    ASYNC_B32(sig_old + fi, wl + O_SIGO + k * 32 + lane);
  }
  {
    int pi = ptBase + lane; if (pi > maxp) pi = maxp;
    ASYNC_B32(p_old + pi, wl + O_POLD + lane);
  }
  ASYNC_WAIT();
  asm volatile("" ::: "memory");

  // ---- Elastic predictor: sig_trial = sig_old + C * (eps - eps_old) ----
  float st[6];
#if HAVE_WMMA_F32X4
  {
    const int half = lane >> 4;   // K/M split across lane halves
    const int lm   = lane & 15;
    const float* le = wl + O_EPS;
    const float* lo = wl + O_EPSO;
#pragma unroll
    for (int t = 0; t < 2; ++t) {           // two 16-point tiles per wave
      const int tp  = t * 16;
      const int row = (tp + lm) * 6;
      const int ks  = 2 * half;             // A lanes 16-31 hold K=2,3
      v2f a0, a1, b0, b1;
      // A (16x4 f32): deps tile, K-step 0 covers comps 0..3
      a0.x = le[row + ks]     - lo[row + ks];
      a0.y = le[row + ks + 1] - lo[row + ks + 1];
      // K-step 1 covers comps 4..7 (6,7 are zero padding)
      a1.x = half ? 0.0f : (le[row + 4] - lo[row + 4]);
      a1.y = half ? 0.0f : (le[row + 5] - lo[row + 5]);
      // B (4x16 f32): rows of the stiffness matrix
      b0.x = cmat(ks, lm);
      b0.y = cmat(ks + 1, lm);
      b1.x = half ? 0.0f : cmat(4, lm);
      b1.y = half ? 0.0f : cmat(5, lm);
      // C accumulator seeded with sig_old in the 16x16 f32 C/D layout
      v8f acc;
#pragma unroll
      for (int r = 0; r < 8; ++r) {
        const int m = tp + r + 8 * half;
        acc[r] = (lm < 6) ? wl[O_SIGO + m * 6 + lm] : 0.0f;
      }
      acc = __builtin_amdgcn_wmma_f32_16x16x4_f32(false, a0, false, b0,
                                                  (short)0, acc, false, false);
      acc = __builtin_amdgcn_wmma_f32_16x16x4_f32(false, a1, false, b1,
                                                  (short)0, acc, false, false);
      // Scatter D (sig_trial) back to LDS, point-major
#pragma unroll
      for (int r = 0; r < 8; ++r) {
        const int m = tp + r + 8 * half;
        if (lm < 6) wl[O_ST + m * 6 + lm] = acc[r];
      }
    }
    asm volatile("" ::: "memory");
#pragma unroll
    for (int j = 0; j < 6; ++j) st[j] = wl[O_ST + lane * 6 + j];
  }
#else
  {
    float dps[6];
#pragma unroll
    for (int j = 0; j < 6; ++j)
      dps[j] = wl[O_EPS + lane * 6 + j] - wl[O_EPSO + lane * 6 + j];
    const float tr = dps[0] + dps[1] + dps[2];
#pragma unroll
    for (int j = 0; j < 6; ++j)
      st[j] = wl[O_SIGO + lane * 6 + j] + ((j < 3) ? kLam * tr : 0.0f)
              + 2.0f * kMu * dps[j];
  }
#endif

  const float p0 = wl[O_POLD + lane];

  // ---- Newton on (dp, depsp) with analytic Jacobian + exact block solve ----
  // J = [[a, b n^T], [-n, M]],  M = I + alpha*D - beta*n n^T
  // M vol = vol ; M n = gamma n ; M r_perp = (1+alpha) r_perp   (tr(n)=0)
  float dp = 0.0f;
  float de[6] = {0.0f, 0.0f, 0.0f, 0.0f, 0.0f, 0.0f};
#pragma unroll
  for (int it = 0; it < kIters; ++it) {
    const float trd = (de[0] + de[1] + de[2]) * (1.0f / 3.0f);
    float sg[6];
#pragma unroll
    for (int j = 0; j < 6; ++j) {
      const float dd = de[j] - ((j < 3) ? trd : 0.0f);
      sg[j] = st[j] - 2.0f * kMu * dd;
    }
    const float trs = (sg[0] + sg[1] + sg[2]) * (1.0f / 3.0f);
    float s[6], ss = 0.0f;
#pragma unroll
    for (int j = 0; j < 6; ++j) {
      s[j] = sg[j] - ((j < 3) ? trs : 0.0f);
      ss += s[j] * s[j];
    }
    const float seq  = sqrtf(1.5f * ss + kTiny);
    const float iseq = 1.0f / seq;
    const float f    = seq - (kSig0 + kH * (p0 + dp));
    float n[6], r2[6], nn = 0.0f, ndr = 0.0f;
#pragma unroll
    for (int j = 0; j < 6; ++j) {
      n[j]  = 1.5f * s[j] * iseq;
      r2[j] = de[j] - n[j] * dp;
      nn  += n[j] * n[j];
      ndr += n[j] * r2[j];
    }
    const float cv  = (r2[0] + r2[1] + r2[2]) * (1.0f / 3.0f);
    const float x   = -f * (1.0f / kE);
    const float rho = sqrtf(x * x + dp * dp + kTiny);
    const float r1  = x + dp - rho;
    const float gx  = 1.0f - x / rho;
    const float gy  = 1.0f - dp / rho;
    const float a   = gx * (kH / kE) + gy;
    const float b   = gx * (2.0f * kMu / kE);
    const float alpha = 3.0f * kMu * dp * iseq;
    const float beta  = 2.0f * kMu * dp * iseq;
    const float gamma = 1.0f + alpha - beta * nn;
    const float ig = 1.0f / gamma;
    const float ia = 1.0f / (1.0f + alpha);
    const float ddp = (r1 - b * ndr * ig) / (a + b * nn * ig);
    const float cn  = ndr / fmaxf(nn, 1e-30f);
#pragma unroll
    for (int j = 0; j < 6; ++j) {
      const float vj = (j < 3) ? cv : 0.0f;
      const float mi = vj + cn * ig * n[j] + (r2[j] - vj - cn * n[j]) * ia;
      de[j] -= mi + ddp * ig * n[j];
    }
    dp -= ddp;
  }

  // ---- Outputs ----
  const int  gi = ptBase + lane;
  if (gi < N) {
    const float trd = (de[0] + de[1] + de[2]) * (1.0f / 3.0f);
#pragma unroll
    for (int j = 0; j < 6; ++j) {
      const float dd = de[j] - ((j < 3) ? trd : 0.0f);
      out_sig[gi * 6 + j]  = st[j] - 2.0f * kMu * dd;
      out_epsp[gi * 6 + j] = epsp_old[gi * 6 + j] + de[j];
    }
    out_p[gi] = p0 + dp;
  }
}

extern "C" void kernel_launch(void* const* d_in, const int* in_sizes, int n_in,
                              void* d_out, int out_size, void* d_ws, size_t ws_size,
                              hipStream_t stream) {
  (void)n_in; (void)out_size; (void)d_ws; (void)ws_size;
  const float* eps      = (const float*)d_in[0];
  const float* eps_old  = (const float*)d_in[1];
  const float* sig_old  = (const float*)d_in[2];
  const float* p_old    = (const float*)d_in[3];
  const float* epsp_old = (const float*)d_in[4];
  const int N = in_sizes[0] / 6;
  float* out      = (float*)d_out;
  float* out_sig  = out;                      // (N,6)
  float* out_p    = out + (size_t)6 * N;      // (N,)
  float* out_epsp = out + (size_t)7 * N;      // (N,6)
  const int blocks = (N + kThreads - 1) / kThreads;
  plasticity_return_map<<<blocks, kThreads, 0, stream>>>(
      eps, eps_old, sig_old, p_old, epsp_old, out_sig, out_p, out_epsp, N);
}